// DiffusionLoss_26319559590071
// MI455X (gfx1250) — compile-verified
//
#include <hip/hip_runtime.h>
#include <math.h>

typedef float v2f __attribute__((ext_vector_type(2)));
typedef float v8f __attribute__((ext_vector_type(8)));

#define WAVES_PER_BLOCK 8
#define ROWS_PER_BLOCK (WAVES_PER_BLOCK * 16)   // 128 rows of p per workgroup
#define NOISE_BLOCKS 256
#define MAX_M 16384
#define UNROLL 4                                // 4 column tiles (64 cols) / iter

// ---------------------------------------------------------------------------
// Kernel 0: init column-min buffer to +inf
// ---------------------------------------------------------------------------
__global__ void init_colmin_kernel(float* __restrict__ colmin, int n) {
    int i = blockIdx.x * blockDim.x + threadIdx.x;
    if (i < n) colmin[i] = __builtin_inff();
}

// ---------------------------------------------------------------------------
// Kernel 1: L1 noise loss partial sums (deterministic fixed-tree reduction)
// ---------------------------------------------------------------------------
__global__ __launch_bounds__(256)
void noise_kernel(const float* __restrict__ a, const float* __restrict__ b,
                  float* __restrict__ partial, int n) {
    __shared__ float s[256];
    float acc = 0.0f;
    for (int i = blockIdx.x * blockDim.x + threadIdx.x; i < n;
         i += gridDim.x * blockDim.x)
        acc += fabsf(a[i] - b[i]);
    s[threadIdx.x] = acc;
    __syncthreads();
    for (int off = 128; off > 0; off >>= 1) {
        if ((int)threadIdx.x < off) s[threadIdx.x] += s[threadIdx.x + off];
        __syncthreads();
    }
    if (threadIdx.x == 0) partial[blockIdx.x] = s[0];
}

// ---------------------------------------------------------------------------
// Kernel 2: chamfer via V_WMMA_F32_16X16X4_F32, software-pipelined.
// Each block: 8 waves x 16 rows = 128 rows of p for batch blockIdx.y.
// Sweeps M columns of t, 4 WMMA tiles (64 cols) per iteration, with next
// iteration's t points loaded before computing the current ones.
//
// The full squared distance comes straight out of the matrix unit:
//   A row i  = (-2 p_x, -2 p_y, -2 p_z, 0)      (K padded from 3 to 4)
//   B col j  = ( t_x,    t_y,    t_z,   0)
//   C[r]     = psq_i + tsq_j
//   D = A*B + C = psq + tsq - 2 p.t  =  d_ij    (no epilogue FMA needed)
//
// rowmin kept in registers; colmin merged via LDS ds_min_num_f32 from all
// 32 lanes (lanes l and l+16 share a column -> the atomic merges halves),
// then global_atomic_min_num_f32 into colmin_g. min is exact => deterministic.
// ---------------------------------------------------------------------------
__global__ __launch_bounds__(256)
void chamfer_kernel(const float* __restrict__ p, const float* __restrict__ t,
                    float* __restrict__ rowvals, float* __restrict__ colmin_g,
                    int N, int M) {
    __shared__ float lds_colmin[MAX_M];

    const int b    = blockIdx.y;
    const int lane = threadIdx.x & 31;
    const int wave = threadIdx.x >> 5;
    const int half = lane >> 4;        // 0: lanes 0-15, 1: lanes 16-31
    const int l16  = lane & 15;

    for (int c = threadIdx.x; c < M; c += blockDim.x)
        lds_colmin[c] = __builtin_inff();
    __syncthreads();

    const float* pb = p + (size_t)b * N * 3;
    const float* tb = t + (size_t)b * M * 3;

    const int rowbase = blockIdx.x * ROWS_PER_BLOCK + wave * 16;

    // Each lane loads p-point (rowbase + l16): covers A fragment + psq source.
    const float px = pb[(rowbase + l16) * 3 + 0];
    const float py = pb[(rowbase + l16) * 3 + 1];
    const float pz = pb[(rowbase + l16) * 3 + 2];
    const float psq_mine = px * px + py * py + pz * pz;

    // A fragment pre-scaled by -2, 16x4 f32 layout:
    // lanes 0-15 hold K=0,1 (-2x,-2y); lanes 16-31 hold K=2,3 (-2z,0)
    v2f afrag;
    afrag.x = half ? (-2.0f * pz) : (-2.0f * px);
    afrag.y = half ? 0.0f         : (-2.0f * py);

    // psq for C/D row layout: VGPR r covers row M = r + 8*half (col = lane%16)
    float psq[8];
#pragma unroll
    for (int r = 0; r < 8; ++r)
        psq[r] = __shfl(psq_mine, r + 8 * half, 32);

    float rowmin[8];
#pragma unroll
    for (int r = 0; r < 8; ++r) rowmin[r] = __builtin_inff();

    // ---- software pipeline: preload first 4 tiles of t ----
    float cx[UNROLL], cy[UNROLL], cz[UNROLL];
#pragma unroll
    for (int u = 0; u < UNROLL; ++u) {
        const int col = u * 16 + l16;
        cx[u] = tb[col * 3 + 0];
        cy[u] = tb[col * 3 + 1];
        cz[u] = tb[col * 3 + 2];
    }

    for (int j = 0; j < M; j += 16 * UNROLL) {
        // issue next group's loads before touching current values
        const int jn = (j + 16 * UNROLL < M) ? (j + 16 * UNROLL) : 0; // wrap, no branch
        float nx[UNROLL], ny[UNROLL], nz[UNROLL];
#pragma unroll
        for (int u = 0; u < UNROLL; ++u) {
            const int col = jn + u * 16 + l16;
            nx[u] = tb[col * 3 + 0];
            ny[u] = tb[col * 3 + 1];
            nz[u] = tb[col * 3 + 2];
        }

#pragma unroll
        for (int u = 0; u < UNROLL; ++u) {
            const float tsq = cx[u] * cx[u] + cy[u] * cy[u] + cz[u] * cz[u];

            // B fragment, 4x16 f32: lanes 0-15 K=0,1 (x,y); lanes 16-31 K=2,3 (z,0)
            v2f bfrag;
            bfrag.x = half ? cz[u] : cx[u];
            bfrag.y = half ? 0.0f : cy[u];

            // accumulator carries psq_i + tsq_j -> D is the squared distance
            v8f cacc;
#pragma unroll
            for (int r = 0; r < 8; ++r) cacc[r] = tsq + psq[r];

            v8f d = __builtin_amdgcn_wmma_f32_16x16x4_f32(
                /*neg_a=*/false, afrag, /*neg_b=*/false, bfrag,
                /*c_mod=*/(short)0, cacc, /*reuse_a=*/false, /*reuse_b=*/false);

            float cm = __builtin_inff();
#pragma unroll
            for (int r = 0; r < 8; ++r) {
                rowmin[r] = fminf(rowmin[r], d[r]);
                cm = fminf(cm, d[r]);
            }
            // all 32 lanes min into the column slot; lanes l and l+16 share a
            // column, the LDS atomic itself merges the two half-partials.
            __hip_atomic_fetch_min(&lds_colmin[j + u * 16 + l16], cm,
                                   __ATOMIC_RELAXED, __HIP_MEMORY_SCOPE_WORKGROUP);
        }

#pragma unroll
        for (int u = 0; u < UNROLL; ++u) {
            cx[u] = nx[u];
            cy[u] = ny[u];
            cz[u] = nz[u];
        }
    }

    // rowmin: reduce across the 16 lanes of each half (columns j ≡ lane mod 16)
#pragma unroll
    for (int mask = 1; mask <= 8; mask <<= 1) {
#pragma unroll
        for (int r = 0; r < 8; ++r)
            rowmin[r] = fminf(rowmin[r], __shfl_xor(rowmin[r], mask, 32));
    }
    if (l16 == 0) {
#pragma unroll
        for (int r = 0; r < 8; ++r) {
            const int row = rowbase + r + 8 * half;
            rowvals[(size_t)b * N + row] = fmaxf(rowmin[r], 0.0f);
        }
    }

    __syncthreads();
    // merge block-local column mins into global (global_atomic_min_num_f32)
    for (int c = threadIdx.x; c < M; c += blockDim.x) {
        __hip_atomic_fetch_min(&colmin_g[(size_t)b * M + c], lds_colmin[c],
                               __ATOMIC_RELAXED, __HIP_MEMORY_SCOPE_AGENT);
    }
}

// ---------------------------------------------------------------------------
// Kernel 3: deterministic finisher -> scalar
// ---------------------------------------------------------------------------
__global__ __launch_bounds__(256)
void final_kernel(const float* __restrict__ noise_partial, int nblocks,
                  const float* __restrict__ rowvals,
                  const float* __restrict__ colmin,
                  int total_bn, float inv_noise_count, float inv_bn,
                  float* __restrict__ out) {
    __shared__ double s[256];
    double acc = 0.0;
    for (int i = threadIdx.x; i < total_bn; i += 256)
        acc += (double)rowvals[i] + (double)fmaxf(colmin[i], 0.0f);
    s[threadIdx.x] = acc;
    __syncthreads();
    for (int off = 128; off > 0; off >>= 1) {
        if ((int)threadIdx.x < off) s[threadIdx.x] += s[threadIdx.x + off];
        __syncthreads();
    }
    if (threadIdx.x == 0) {
        double ns = 0.0;
        for (int i = 0; i < nblocks; ++i) ns += (double)noise_partial[i];
        const double noise_loss   = ns * (double)inv_noise_count;
        const double chamfer_mean = s[0] * (double)inv_bn;
        out[0] = (float)(noise_loss + 0.1 * chamfer_mean);
    }
}

// ---------------------------------------------------------------------------
extern "C" void kernel_launch(void* const* d_in, const int* in_sizes, int n_in,
                              void* d_out, int out_size, void* d_ws, size_t ws_size,
                              hipStream_t stream) {
    const float* pred_noise   = (const float*)d_in[0];
    const float* actual_noise = (const float*)d_in[1];
    const float* pred_pts     = (const float*)d_in[2];
    const float* target_pts   = (const float*)d_in[3];
    float* out = (float*)d_out;

    const int B = 8;
    const int N = in_sizes[2] / (B * 3);   // 16384
    const int M = in_sizes[3] / (B * 3);   // 16384
    const int noise_n = in_sizes[0];       // B*N*3

    // workspace layout (floats)
    float* colmin        = (float*)d_ws;                // B*M
    float* rowvals       = colmin + (size_t)B * M;      // B*N
    float* noise_partial = rowvals + (size_t)B * N;     // NOISE_BLOCKS

    // 0) colmin := +inf
    {
        const int n = B * M;
        init_colmin_kernel<<<(n + 255) / 256, 256, 0, stream>>>(colmin, n);
    }
    // 1) noise L1 partials
    noise_kernel<<<NOISE_BLOCKS, 256, 0, stream>>>(pred_noise, actual_noise,
                                                   noise_partial, noise_n);
    // 2) chamfer (WMMA f32 16x16x4, pipelined x4, distance folded into C)
    {
        dim3 grid(N / ROWS_PER_BLOCK, B);
        chamfer_kernel<<<grid, 256, 0, stream>>>(pred_pts, target_pts,
                                                 rowvals, colmin, N, M);
    }
    // 3) combine to scalar
    final_kernel<<<1, 256, 0, stream>>>(noise_partial, NOISE_BLOCKS,
                                        rowvals, colmin, B * N,
                                        1.0f / (float)noise_n,
                                        1.0f / (float)(B * N), out);
}